// MultipleNegativesRankingLoss_57501022159572
// MI455X (gfx1250) — compile-verified
//
#include <hip/hip_runtime.h>
#include <math.h>

// Problem dims (fixed by setup_inputs)
#define Q 32
#define L 128
#define D 128
#define N 64
#define SCALE 20.0f

typedef __attribute__((ext_vector_type(2))) float v2f;
typedef __attribute__((ext_vector_type(8))) float v8f;

// ---------------- Stage 0: zero accumulators ----------------
__global__ void zero_kernel(float* __restrict__ p, int n) {
    int i = blockIdx.x * blockDim.x + threadIdx.x;
    if (i < n) p[i] = 0.0f;
}

// ---------------- Stage A: row norms (rows x D, D=128) ----------------
__global__ void norms_kernel(const float* __restrict__ x, float* __restrict__ nrm, int rows) {
    int i = blockIdx.x * blockDim.x + threadIdx.x;
    if (i >= rows) return;
    const float4* p = (const float4*)(x + (size_t)i * D);
    float s = 0.0f;
#pragma unroll 8
    for (int k = 0; k < D / 4; ++k) {
        float4 v = p[k];
        s += v.x * v.x + v.y * v.y + v.z * v.z + v.w * v.w;
    }
    nrm[i] = s;
}

// ---------------- Stage B: per-q mask counts, clipped >= 1 ----------------
__global__ __launch_bounds__(L) void counts_kernel(const int* __restrict__ mask,
                                                   float* __restrict__ counts) {
    int q = blockIdx.x;
    int t = threadIdx.x;
    float v = (float)mask[q * L + t];
#pragma unroll
    for (int off = 16; off > 0; off >>= 1) v += __shfl_xor(v, off);
    __shared__ float sm[L / 32];
    if ((t & 31) == 0) sm[t >> 5] = v;
    __syncthreads();
    if (t == 0) {
        float s = 0.0f;
#pragma unroll
        for (int w = 0; w < L / 32; ++w) s += sm[w];
        counts[q] = fmaxf(s, 1.0f);
    }
}

// FP32 WMMA step: D(16x16,f32) += A(16x4,f32) * B(4x16,f32), one wave.
__device__ __forceinline__ v8f wmma_f32_step(v2f a, v2f b, v8f c) {
    // 8 args: (neg_a, A, neg_b, B, c_mod, C, reuse_a, reuse_b)
    return __builtin_amdgcn_wmma_f32_16x16x4_f32(false, a, false, b, (short)0, c, false, false);
}

// ---------------- Stage C: dot(x, y^T) tiles -> masked-dist column sums ----
// grid = (Q*L/16, N/16), block = 32 (one wave per 16x16 tile)
__global__ __launch_bounds__(32) void dot_xy_kernel(const float* __restrict__ x,
                                                    const float* __restrict__ y,
                                                    const int*   __restrict__ mask,
                                                    const float* __restrict__ normx,
                                                    const float* __restrict__ normy,
                                                    float*       __restrict__ eddot) {
    const int lane = threadIdx.x;
    const int m    = lane & 15;   // row (A) / col (B) within tile
    const int kh   = lane >> 4;   // K half-select (K pairs {0,1} vs {2,3})
    const int row0 = blockIdx.x * 16;        // global row in [0, Q*L)
    const int n0   = blockIdx.y * 16;        // global col in [0, N)
    const int q    = row0 / L;               // 16-row tile never crosses q (L%16==0)

    const float* Ap = x + (size_t)(row0 + m) * D + 2 * kh;
    const float* Bp = y + (size_t)(n0 + m) * D + 2 * kh;   // B[k][n] = y[n][k]

    v8f c = {};
#pragma unroll 8
    for (int k = 0; k < D / 4; ++k) {
        v2f a = *(const v2f*)(Ap + 4 * k);
        v2f b = *(const v2f*)(Bp + 4 * k);
        c = wmma_f32_step(a, b, c);
    }

    // epilogue: dist = sqrt(max(nx + ny - 2*dot, 0)) * mask_row, sum over 16 rows
    const float ny = normy[n0 + m];
    float s = 0.0f;
#pragma unroll
    for (int v = 0; v < 8; ++v) {
        int row   = row0 + v + 8 * kh;       // C/D layout: M = v (+8 for upper half)
        float sq  = normx[row] + ny - 2.0f * c[v];
        sq        = fmaxf(sq, 0.0f);
        s += sqrtf(sq) * (float)mask[row];
    }
    s += __shfl_xor(s, 16);                  // fold the two half-wave row groups
    if (lane < 16) atomicAdd(&eddot[q * N + n0 + m], s);
}

// ---------------- Stage D: per-q Gram tiles -> ed_q sums --------------------
// grid = (L/16, L/16, Q), block = 32
__global__ __launch_bounds__(32) void gram_kernel(const float* __restrict__ x,
                                                  const int*   __restrict__ mask,
                                                  const float* __restrict__ normx,
                                                  float*       __restrict__ edq) {
    const int lane = threadIdx.x;
    const int m    = lane & 15;
    const int kh   = lane >> 4;
    const int i0   = blockIdx.x * 16;
    const int j0   = blockIdx.y * 16;
    const int q    = blockIdx.z;

    const float* Ap = x + ((size_t)q * L + i0 + m) * D + 2 * kh;
    const float* Bp = x + ((size_t)q * L + j0 + m) * D + 2 * kh;  // B[k][j] = x[q][j][k]

    v8f c = {};
#pragma unroll 8
    for (int k = 0; k < D / 4; ++k) {
        v2f a = *(const v2f*)(Ap + 4 * k);
        v2f b = *(const v2f*)(Bp + 4 * k);
        c = wmma_f32_step(a, b, c);
    }

    const int   jcol = q * L + j0 + m;
    const float nj   = normx[jcol];
    const float mj   = (float)mask[jcol];
    float s = 0.0f;
#pragma unroll
    for (int v = 0; v < 8; ++v) {
        int irow  = q * L + i0 + v + 8 * kh;
        float sq  = normx[irow] + nj - 2.0f * c[v];
        sq        = fmaxf(sq, 0.0f);
        s += sqrtf(sq) * (float)mask[irow] * mj;
    }
#pragma unroll
    for (int off = 16; off > 0; off >>= 1) s += __shfl_xor(s, off);
    if (lane == 0) atomicAdd(&edq[q], s);
}

// ---------------- Stage E: scores + logsumexp + mean loss -------------------
// one block, 32 waves; wave w handles q = w (N=64 -> 2 scores per lane)
__global__ __launch_bounds__(1024) void loss_kernel(const float* __restrict__ eddot,
                                                    const float* __restrict__ edq_sum,
                                                    const float* __restrict__ counts,
                                                    float*       __restrict__ out) {
    const int q    = threadIdx.x >> 5;
    const int lane = threadIdx.x & 31;

    float cq    = counts[q];                   // already clipped >= 1
    float valid = fmaxf(cq * cq, 1.0f);
    float edq   = edq_sum[q] / valid;

    // scores = (2*ed_sums - ed_q) * SCALE  (ed_sums = eddot / counts)
    float s0 = (2.0f * eddot[q * N + lane]      / cq - edq) * SCALE;
    float s1 = (2.0f * eddot[q * N + lane + 32] / cq - edq) * SCALE;

    float mx = fmaxf(s0, s1);
#pragma unroll
    for (int off = 16; off > 0; off >>= 1) mx = fmaxf(mx, __shfl_xor(mx, off));
    float e = expf(s0 - mx) + expf(s1 - mx);
#pragma unroll
    for (int off = 16; off > 0; off >>= 1) e += __shfl_xor(e, off);
    float logz = mx + logf(e);

    float diag = __shfl(s0, q);                // q < 32 -> label column is in s0 of lane q
    float r = logz - diag;

    __shared__ float sm[Q];
    if (lane == 0) sm[q] = r;
    __syncthreads();
    if (threadIdx.x == 0) {
        float acc = 0.0f;
#pragma unroll
        for (int i = 0; i < Q; ++i) acc += sm[i];
        out[0] = acc / (float)Q;
    }
}

extern "C" void kernel_launch(void* const* d_in, const int* in_sizes, int n_in,
                              void* d_out, int out_size, void* d_ws, size_t ws_size,
                              hipStream_t stream) {
    const float* x    = (const float*)d_in[0];  // anchors   [Q,L,D]
    const float* y    = (const float*)d_in[1];  // candidates[N,D]
    const int*   mask = (const int*)d_in[2];    // attention_mask [Q,L]
    float* out = (float*)d_out;

    float* ws     = (float*)d_ws;
    float* normx  = ws;            // Q*L   = 4096
    float* normy  = ws + 4096;     // N     = 64
    float* counts = ws + 4160;     // Q     = 32
    float* eddot  = ws + 4192;     // Q*N   = 2048  (atomic accum)
    float* edq    = ws + 6240;     // Q     = 32    (atomic accum, contiguous w/ eddot)

    // zero the atomic accumulators (eddot + edq contiguous: 2080 floats)
    zero_kernel<<<(2080 + 255) / 256, 256, 0, stream>>>(eddot, 2080);

    // norms + counts
    norms_kernel<<<(Q * L + 255) / 256, 256, 0, stream>>>(x, normx, Q * L);
    norms_kernel<<<1, 64, 0, stream>>>(y, normy, N);
    counts_kernel<<<Q, L, 0, stream>>>(mask, counts);

    // cross term x.y^T  (256 row-tiles x 4 col-tiles, one wave each)
    dim3 gC(Q * L / 16, N / 16);
    dot_xy_kernel<<<gC, 32, 0, stream>>>(x, y, mask, normx, normy, eddot);

    // per-q Gram matrices (8 x 8 tiles per q)
    dim3 gD(L / 16, L / 16, Q);
    gram_kernel<<<gD, 32, 0, stream>>>(x, mask, normx, edq);

    // final scores + CE loss
    loss_kernel<<<1, 1024, 0, stream>>>(eddot, edq, counts, out);
}